// MoE_lora_14242111553983
// MI455X (gfx1250) — compile-verified
//
#include <hip/hip_runtime.h>
#include <math.h>

typedef __attribute__((ext_vector_type(2))) float v2f;
typedef __attribute__((ext_vector_type(8))) float v8f;

#define NB  64
#define NL  256
#define DIN 768
#define NH  64
#define NE  8
#define BLH (NB*NL*NH)

__device__ __forceinline__ float gelu_f(float x) {
    return 0.5f * x * (1.0f + erff(x * 0.70710678118654752f));
}

__device__ __forceinline__ v8f wmma4(v2f a, v2f b, v8f c) {
    return __builtin_amdgcn_wmma_f32_16x16x4_f32(false, a, false, b, (short)0, c, false, false);
}

// A fragment (16x4 f32) from a 16x64 row-major LDS tile.
// lanes 0-15: K=kk,kk+1 ; lanes 16-31: K=kk+2,kk+3 (ISA 7.12.2)
__device__ __forceinline__ v2f ldsAfrag(const float* As, int kk, int lane) {
    int half = lane >> 4, m = lane & 15;
    v2f a; a.x = As[m*64 + kk + 2*half]; a.y = As[m*64 + kk + 2*half + 1]; return a;
}
__device__ __forceinline__ v2f ldsAfrag_gelu(const float* As, int kk, int lane) {
    int half = lane >> 4, m = lane & 15;
    v2f a; a.x = gelu_f(As[m*64 + kk + 2*half]); a.y = gelu_f(As[m*64 + kk + 2*half + 1]); return a;
}
// B fragment (4x16 f32) directly from a row-major global weight matrix (ldb = N).
__device__ __forceinline__ v2f gBfrag(const float* W, int ldb, int kk, int n, int lane) {
    int half = lane >> 4;
    int r = kk + 2*half;
    v2f b; b.x = W[r*ldb + n]; b.y = W[(r+1)*ldb + n]; return b;
}
// Full K=64 GEMM accumulate: 16x16 output fragment
__device__ __forceinline__ v8f gemmK64(const float* As, const float* W, int n, int lane, v8f acc) {
#pragma unroll
    for (int kk = 0; kk < 64; kk += 4)
        acc = wmma4(ldsAfrag(As, kk, lane), gBfrag(W, 64, kk, n, lane), acc);
    return acc;
}

// Async global->LDS copy of one 16x64 f32 tile (strided rows), 4 waves cooperating:
// each wave issues two B128 ops (32 lanes x 16B = 512B each).
__device__ __forceinline__ void issue_async_tile(const float* gsrc, float* lds,
                                                 int m0, int k0, int wid, int lane) {
#pragma unroll
    for (int j = 0; j < 2; ++j) {
        int f = ((wid * 2 + j) * 32 + lane) * 4;      // float index, 16B granules
        int r = f >> 6, c = f & 63;
        unsigned go = (unsigned)((((m0 + r) * DIN) + k0 + c) * 4);
        asm volatile("global_load_async_to_lds_b128 %0, %1, %2"
                     :: "v"((unsigned)(size_t)&lds[f]), "v"(go), "s"(gsrc) : "memory");
    }
}

// ---------------- Kernel 1: per-row mean of dte over DIN ----------------
__global__ void k_rowmean(const float* __restrict__ dte, float* __restrict__ rowmean) {
    int wid = threadIdx.x >> 5, lane = threadIdx.x & 31;
    int row = blockIdx.x * 8 + wid;            // 2048 blocks * 8 waves = 16384 rows
    const float* p = dte + (size_t)row * DIN;
    float s = 0.f;
    for (int d = lane; d < DIN; d += 32) s += p[d];
#pragma unroll
    for (int m = 16; m > 0; m >>= 1) s += __shfl_xor(s, m, 32);
    if (lane == 0) rowmean[row] = s * (1.0f / DIN);
}

// ---------------- Kernel 2: gating (logits, top2, softmax, loss, gb) ----------------
__global__ void k_gate(const float* __restrict__ rowmean, const float* __restrict__ w_gate,
                       const float* __restrict__ b_exp, float* __restrict__ gpack,
                       float* __restrict__ gb, float* __restrict__ loss_out) {
    __shared__ float logitS[NB][NE];
    __shared__ float gateS[NB][NE];
    __shared__ float impS[NE], ldS[NE];
    __shared__ float pk[NB][4];
    int t = threadIdx.x;                        // blockDim = 512
    {
        int b = t >> 3, e = t & 7;
        const float* rm = rowmean + b * NL;
        float s = 0.f;
        for (int l = 0; l < NL; ++l) s += rm[l] * w_gate[l*NE + e];
        logitS[b][e] = s;
        gateS[b][e] = 0.f;
    }
    __syncthreads();
    if (t < NB) {
        int b = t;
        int e0 = 0; float v0 = logitS[b][0];
        for (int e = 1; e < NE; ++e) { float v = logitS[b][e]; if (v > v0) { v0 = v; e0 = e; } }
        int e1 = (e0 == 0) ? 1 : 0; float v1 = logitS[b][e1];
        for (int e = 0; e < NE; ++e) {
            if (e == e0 || e == e1) continue;
            float v = logitS[b][e]; if (v > v1) { v1 = v; e1 = e; }
        }
        float x0 = expf(v0 - v0), x1 = expf(v1 - v0);
        float inv = 1.0f / (x0 + x1);
        float g0 = x0 * inv, g1 = x1 * inv;
        gateS[b][e0] = g0; gateS[b][e1] = g1;
        pk[b][0] = g0; pk[b][1] = g1;
        pk[b][2] = __int_as_float(e0); pk[b][3] = __int_as_float(e1);
        gpack[b*4+0] = g0; gpack[b*4+1] = g1;
        gpack[b*4+2] = __int_as_float(e0); gpack[b*4+3] = __int_as_float(e1);
    }
    __syncthreads();
    if (t < NE) {
        float si = 0.f, sl = 0.f;
        for (int b = 0; b < NB; ++b) { float g = gateS[b][t]; si += g; sl += (g > 0.f) ? 1.f : 0.f; }
        impS[t] = si; ldS[t] = sl;
    }
    __syncthreads();
    if (t == 0) {
        float mi = 0.f, ml = 0.f;
        for (int e = 0; e < NE; ++e) { mi += impS[e]; ml += ldS[e]; }
        mi *= (1.0f/NE); ml *= (1.0f/NE);
        float vi = 0.f, vl = 0.f;
        for (int e = 0; e < NE; ++e) {
            float di = impS[e]-mi, dl = ldS[e]-ml; vi += di*di; vl += dl*dl;
        }
        vi *= (1.0f/(NE-1)); vl *= (1.0f/(NE-1));   // ddof=1 per cv_squared
        loss_out[0] = 0.01f * (vi/(mi*mi + 1e-10f) + vl/(ml*ml + 1e-10f));
    }
    for (int i = t; i < NB*NH; i += blockDim.x) {   // gb = gates @ b_exp
        int b = i >> 6, h = i & 63;
        int e0 = __float_as_int(pk[b][2]), e1 = __float_as_int(pk[b][3]);
        gb[i] = pk[b][0]*b_exp[e0*NH + h] + pk[b][1]*b_exp[e1*NH + h];
    }
}

// ---------------- Kernel 3: fused big GEMMs (sh,y,rgb) + LN + W_px ----------------
__global__ void __launch_bounds__(128) k_front(
    const float* __restrict__ x, const float* __restrict__ dte,
    const float* __restrict__ W_exp, const float* __restrict__ W_sh, const float* __restrict__ b_sh,
    const float* __restrict__ W_rgb, const float* __restrict__ b_rgb,
    const float* __restrict__ ln_g, const float* __restrict__ ln_b,
    const float* __restrict__ W_px, const float* __restrict__ b_px,
    const float* __restrict__ gpack, const float* __restrict__ gb,
    float* __restrict__ sh_o, float* __restrict__ y_o, float* __restrict__ rgb_o,
    float* __restrict__ t1_o, float* __restrict__ t2_o)
{
    __shared__ float Ad[2][16*64];              // double-buffered async-DMA A tiles
    __shared__ float Ax[2][16*64];
    __shared__ float Sh[16*64];
    __shared__ float T [16*64];
    int tid = threadIdx.x, lane = tid & 31, wid = tid >> 5;
    int m0 = blockIdx.x * 16;                   // 16-row tile (never crosses batch: 256 rows/batch)
    int b  = m0 >> 8;
    float g0 = gpack[b*4+0], g1 = gpack[b*4+1];
    int e0 = __float_as_int(gpack[b*4+2]), e1 = __float_as_int(gpack[b*4+3]);
    const float* We0 = W_exp + (size_t)e0 * DIN * NH;
    const float* We1 = W_exp + (size_t)e1 * DIN * NH;
    int nl = lane & 15, half = lane >> 4;
    int n = wid * 16 + nl;
    v8f acc_sh = {}; v8f acc_y = {}; v8f acc_rgb = {};

    // Prologue: DMA chunk 0 into buffer 0 (ASYNCcnt-tracked)
    issue_async_tile(dte, Ad[0], m0, 0, wid, lane);
    issue_async_tile(x,   Ax[0], m0, 0, wid, lane);

    for (int ch = 0; ch < 12; ++ch) {
        int k0 = ch * 64;
        const float* Adc = Ad[ch & 1];
        const float* Axc = Ax[ch & 1];
        if (ch < 11) {
            // Kick off next chunk's DMA, then wait only for the current chunk's
            // 4 (per-wave, in-order) async ops; next chunk overlaps the WMMAs.
            issue_async_tile(dte, Ad[(ch + 1) & 1], m0, k0 + 64, wid, lane);
            issue_async_tile(x,   Ax[(ch + 1) & 1], m0, k0 + 64, wid, lane);
            asm volatile("s_wait_asynccnt 0x4" ::: "memory");
        } else {
            asm volatile("s_wait_asynccnt 0x0" ::: "memory");
        }
        __syncthreads();
#pragma unroll
        for (int kk = 0; kk < 64; kk += 4) {
            v2f ad = ldsAfrag(Adc, kk, lane);
            v2f ax = ldsAfrag(Axc, kk, lane);
            int rB = k0 + kk + 2*half;
            v2f bsh; bsh.x = W_sh[rB*NH + n];            bsh.y = W_sh[(rB+1)*NH + n];
            v2f bmx; bmx.x = g0*We0[rB*NH+n]     + g1*We1[rB*NH+n];
                     bmx.y = g0*We0[(rB+1)*NH+n] + g1*We1[(rB+1)*NH+n];
            v2f brg; brg.x = W_rgb[rB*NH + n];           brg.y = W_rgb[(rB+1)*NH + n];
            acc_sh  = wmma4(ad, bsh, acc_sh);
            acc_y   = wmma4(ad, bmx, acc_y);
            acc_rgb = wmma4(ax, brg, acc_rgb);
        }
        __syncthreads();
    }
    float bs = b_sh[n], br = b_rgb[n], bg = gb[b*NH + n];
#pragma unroll
    for (int r = 0; r < 8; ++r) {
        int m = r + 8*half;
        size_t o = (size_t)(m0 + m)*NH + n;
        float vsh = acc_sh[r] + bs;
        sh_o[o] = vsh;  Sh[m*64 + n] = vsh;
        y_o[o]  = acc_y[r]  + bg;
        rgb_o[o]= acc_rgb[r]+ br;
    }
    __syncthreads();
    if (tid < 16) {                             // LayerNorm over H (ddof=0)
        int m = tid;
        float s = 0.f;
        for (int h = 0; h < NH; ++h) s += Sh[m*64 + h];
        float mean = s * (1.0f/NH);
        float v = 0.f;
        for (int h = 0; h < NH; ++h) { float d = Sh[m*64+h] - mean; v += d*d; }
        float rstd = rsqrtf(v * (1.0f/NH) + 1e-6f);
        for (int h = 0; h < NH; ++h)
            T[m*64 + h] = (Sh[m*64+h] - mean) * rstd * ln_g[h] + ln_b[h];
    }
    __syncthreads();
    // t = T @ W_px (64x128) + b_px ; wave w owns cols [32w, 32w+32)
    v8f accp0 = {}; v8f accp1 = {};
    int c0 = wid*32 + nl, c1 = c0 + 16;
#pragma unroll
    for (int kk = 0; kk < 64; kk += 4) {
        v2f a = ldsAfrag(T, kk, lane);
        accp0 = wmma4(a, gBfrag(W_px, 128, kk, c0, lane), accp0);
        accp1 = wmma4(a, gBfrag(W_px, 128, kk, c1, lane), accp1);
    }
    float bp0 = b_px[c0], bp1 = b_px[c1];
#pragma unroll
    for (int r = 0; r < 8; ++r) {
        int m = r + 8*half;
        size_t rowo = (size_t)(m0 + m)*NH;
        float v0 = accp0[r] + bp0, v1 = accp1[r] + bp1;
        if (c0 < 64) t1_o[rowo + c0] = v0; else t2_o[rowo + c0 - 64] = v0;
        if (c1 < 64) t1_o[rowo + c1] = v1; else t2_o[rowo + c1 - 64] = v1;
    }
}

// ---------------- Kernel 4: depthwise 3x3 conv on 16x16 grid + gelu*t2 ----------------
__global__ void k_conv(const float* __restrict__ t1, const float* __restrict__ t2,
                       const float* __restrict__ csh, float* __restrict__ gt) {
    int idx = blockIdx.x * 256 + threadIdx.x;   // 4096 blocks * 256 = B*L*H
    int h = idx & 63;
    int l = (idx >> 6) & 255;
    int bb = idx >> 14;
    int s1 = l >> 4, s2 = l & 15;
    const float* base = t1 + (size_t)bb * NL * NH;
    float acc = 0.f;
#pragma unroll
    for (int i = 0; i < 3; ++i) {
        int r1 = s1 + i - 1;
        if (r1 < 0 || r1 > 15) continue;
#pragma unroll
        for (int j = 0; j < 3; ++j) {
            int r2 = s2 + j - 1;
            if (r2 < 0 || r2 > 15) continue;
            acc += csh[h*9 + i*3 + j] * base[(r1*16 + r2)*NH + h];
        }
    }
    gt[idx] = gelu_f(acc) * t2[idx];
}

// ---------------- Kernel 5: fused back-end chain of K=64 WMMA GEMMs ----------------
__global__ void __launch_bounds__(128) k_back(
    const float* __restrict__ gt, const float* __restrict__ sh, const float* __restrict__ y,
    const float* __restrict__ rgb,
    const float* __restrict__ W_pxx, const float* __restrict__ b_pxx,
    const float* __restrict__ W_dte, const float* __restrict__ b_dte,
    const float* __restrict__ W_dta, const float* __restrict__ b_dta,
    const float* __restrict__ W_fx,  const float* __restrict__ b_fx,
    const float* __restrict__ W_fmod,const float* __restrict__ b_fmod,
    const float* __restrict__ W_fxx, const float* __restrict__ b_fxx,
    float* __restrict__ out)
{
    __shared__ float GT [16*64];   // reused as M = gelu(dte_tok)
    __shared__ float SHr[16*64];
    __shared__ float Yt [16*64];   // reused as P = gelu(m)*a
    __shared__ float RGB[16*64];
    __shared__ float SH2[16*64];
    int tid = threadIdx.x, lane = tid & 31, wid = tid >> 5;
    int m0 = blockIdx.x * 16;
    int nl = lane & 15, half = lane >> 4;
    int n = wid * 16 + nl;
    {   // tiles are contiguous 4KB spans (row stride == NH): async DMA them in
        unsigned gbase = (unsigned)(m0 * NH * 4);
#pragma unroll
        for (int i = 0; i < 2; ++i) {
            int f = (i * 128 + tid) * 4;
            unsigned go = gbase + (unsigned)(f * 4);
            asm volatile("global_load_async_to_lds_b128 %0, %1, %2"
                         :: "v"((unsigned)(size_t)&GT [f]), "v"(go), "s"(gt)  : "memory");
            asm volatile("global_load_async_to_lds_b128 %0, %1, %2"
                         :: "v"((unsigned)(size_t)&SHr[f]), "v"(go), "s"(sh)  : "memory");
            asm volatile("global_load_async_to_lds_b128 %0, %1, %2"
                         :: "v"((unsigned)(size_t)&Yt [f]), "v"(go), "s"(y)   : "memory");
            asm volatile("global_load_async_to_lds_b128 %0, %1, %2"
                         :: "v"((unsigned)(size_t)&RGB[f]), "v"(go), "s"(rgb) : "memory");
        }
        asm volatile("s_wait_asynccnt 0x0" ::: "memory");
    }
    __syncthreads();
    {   // sh2 = gt @ W_pxx + b_pxx + sh
        v8f acc = {};
        acc = gemmK64(GT, W_pxx, n, lane, acc);
        float bb = b_pxx[n];
#pragma unroll
        for (int r = 0; r < 8; ++r) {
            int m = r + 8*half;
            SH2[m*64 + n] = acc[r] + bb + SHr[m*64 + n];
        }
    }
    __syncthreads();
    {   // dte_tok = y@W_dte + sh2@W_dteall + biases ; M = gelu(dte_tok)
        v8f acc = {};
        acc = gemmK64(Yt,  W_dte, n, lane, acc);
        acc = gemmK64(SH2, W_dta, n, lane, acc);
        float bb = b_dte[n] + b_dta[n];
#pragma unroll
        for (int r = 0; r < 8; ++r) {
            int m = r + 8*half;
            GT[m*64 + n] = gelu_f(acc[r] + bb);
        }
    }
    __syncthreads();
    {   // m = M@W_fmod + b_fmod ; a = gelu(rgb)@W_fx + b_fx ; P = gelu(m)*a
        v8f accm = {}; v8f acca = {};
        accm = gemmK64(GT, W_fmod, n, lane, accm);
#pragma unroll
        for (int kk = 0; kk < 64; kk += 4)
            acca = wmma4(ldsAfrag_gelu(RGB, kk, lane), gBfrag(W_fx, 64, kk, n, lane), acca);
        float bm = b_fmod[n], ba = b_fx[n];
#pragma unroll
        for (int r = 0; r < 8; ++r) {
            int m = r + 8*half;
            Yt[m*64 + n] = gelu_f(accm[r] + bm) * (acca[r] + ba);
        }
    }
    __syncthreads();
    {   // out = P @ W_fxx + b_fxx + rgb
        v8f acc = {};
        acc = gemmK64(Yt, W_fxx, n, lane, acc);
        float bb = b_fxx[n];
#pragma unroll
        for (int r = 0; r < 8; ++r) {
            int m = r + 8*half;
            out[(size_t)(m0 + m)*NH + n] = acc[r] + bb + RGB[m*64 + n];
        }
    }
}

extern "C" void kernel_launch(void* const* d_in, const int* in_sizes, int n_in,
                              void* d_out, int out_size, void* d_ws, size_t ws_size,
                              hipStream_t stream) {
    (void)in_sizes; (void)n_in; (void)out_size; (void)ws_size;
    const float* x      = (const float*)d_in[0];
    const float* dte    = (const float*)d_in[1];
    const float* w_gate = (const float*)d_in[2];
    const float* W_exp  = (const float*)d_in[3];
    const float* b_exp  = (const float*)d_in[4];
    const float* W_sh   = (const float*)d_in[5];
    const float* b_sh   = (const float*)d_in[6];
    const float* ln_g   = (const float*)d_in[7];
    const float* ln_b   = (const float*)d_in[8];
    const float* W_px   = (const float*)d_in[9];
    const float* b_px   = (const float*)d_in[10];
    const float* conv   = (const float*)d_in[11];
    const float* W_pxx  = (const float*)d_in[12];
    const float* b_pxx  = (const float*)d_in[13];
    const float* W_dte  = (const float*)d_in[14];
    const float* b_dte  = (const float*)d_in[15];
    const float* W_dta  = (const float*)d_in[16];
    const float* b_dta  = (const float*)d_in[17];
    const float* W_rgb  = (const float*)d_in[18];
    const float* b_rgb  = (const float*)d_in[19];
    const float* W_fx   = (const float*)d_in[20];
    const float* b_fx   = (const float*)d_in[21];
    const float* W_fmod = (const float*)d_in[22];
    const float* b_fmod = (const float*)d_in[23];
    const float* W_fxx  = (const float*)d_in[24];
    const float* b_fxx  = (const float*)d_in[25];

    float* out = (float*)d_out;
    float* ws  = (float*)d_ws;
    float* rowmean = ws;                 // 16384
    float* gpack   = ws + 16384;         // 256  (g0,g1,e0bits,e1bits per batch)
    float* gb      = ws + 16640;         // 4096
    float* sh_w    = ws + 20736;         // 6 * BLH workspace tensors
    float* y_w     = sh_w  + BLH;
    float* rgb_w   = y_w   + BLH;
    float* t1_w    = rgb_w + BLH;
    float* t2_w    = t1_w  + BLH;
    float* gt_w    = t2_w  + BLH;

    k_rowmean<<<2048, 256, 0, stream>>>(dte, rowmean);
    k_gate<<<1, 512, 0, stream>>>(rowmean, w_gate, b_exp, gpack, gb, out + BLH);
    k_front<<<1024, 128, 0, stream>>>(x, dte, W_exp, W_sh, b_sh, W_rgb, b_rgb,
                                      ln_g, ln_b, W_px, b_px, gpack, gb,
                                      sh_w, y_w, rgb_w, t1_w, t2_w);
    k_conv<<<4096, 256, 0, stream>>>(t1_w, t2_w, conv, gt_w);
    k_back<<<1024, 128, 0, stream>>>(gt_w, sh_w, y_w, rgb_w,
                                     W_pxx, b_pxx, W_dte, b_dte, W_dta, b_dta,
                                     W_fx, b_fx, W_fmod, b_fmod, W_fxx, b_fxx, out);
}